// MPSOpen_16527034155480
// MI455X (gfx1250) — compile-verified
//
#include <hip/hip_runtime.h>
#include <hip/hip_bf16.h>

typedef float v2f __attribute__((ext_vector_type(2)));
typedef float v8f __attribute__((ext_vector_type(8)));
typedef int   v4i __attribute__((ext_vector_type(4)));

typedef __attribute__((address_space(1))) v4i* gv4i_p;   // global
typedef __attribute__((address_space(3))) v4i* lv4i_p;   // LDS

namespace {
constexpr int kB      = 4096;
constexpr int kL      = 128;
constexpr int kD      = 64;          // bond dimension
constexpr int kMid    = kL - 2;      // 126 middle sites
constexpr int kWaves  = 4;           // waves per block (wave32)
constexpr int kTile   = 16;          // batch rows per wave (WMMA M)
constexpr int kThreads = kWaves * 32;                 // 128
constexpr int kBlocks  = kB / (kWaves * kTile);       // 64
constexpr int kSiteFloats = 2 * kD * kD;              // 8192 floats = 32 KB / site
}

// ---- CDNA5 async global->LDS copy (ASYNCcnt-tracked, no VGPR round trip) ----
__device__ __forceinline__ void async_copy_b128(const void* g, void* l) {
#if __has_builtin(__builtin_amdgcn_global_load_async_to_lds_b128)
  __builtin_amdgcn_global_load_async_to_lds_b128((gv4i_p)g, (lv4i_p)l, 0, 0);
#else
  const unsigned lds_off = (unsigned)(size_t)(lv4i_p)l;
  asm volatile("global_load_async_to_lds_b128 %0, %1, off"
               :: "v"(lds_off), "v"((unsigned long long)(size_t)g)
               : "memory");
#endif
}

__device__ __forceinline__ void wait_async0() {
#if __has_builtin(__builtin_amdgcn_s_wait_asynccnt)
  __builtin_amdgcn_s_wait_asynccnt(0);
#else
  asm volatile("s_wait_asynccnt 0x0" ::: "memory");
#endif
}

// Stage both q-matrices of one site (32 KB) into an LDS buffer, asynchronously.
// 128 threads x b128(16B) = 2 KB per step -> 16 steps.
__device__ __forceinline__ void stage_site(const float* __restrict__ midT, int s,
                                           float* lbuf, int tid) {
  const char* g = (const char*)(midT + (size_t)s * kSiteFloats);
  char* l = (char*)lbuf;
#pragma unroll
  for (int i = 0; i < (kSiteFloats * 4) / (16 * kThreads); ++i) {
    const int off = (i * kThreads + tid) * 16;
    async_copy_b128(g + off, l + off);
  }
}

__global__ __launch_bounds__(kThreads)
void mps_logpsi_kernel(const int*   __restrict__ qn,      // [B, L]
                       const float* __restrict__ leftT,   // [d, D]
                       const float* __restrict__ rightT,  // [d, D]
                       const float* __restrict__ midT,    // [L-2, d, D, D]
                       float*       __restrict__ out)     // [B] complex64 -> 2*B floats
{
  // Double-buffered site matrices (both q values): 2 x 32 KB
  __shared__ float smM[2][kSiteFloats];
  // Per-wave edge matrix E[16][64], row-major: 4 x 4 KB
  __shared__ float smE[kWaves][kTile * kD];

  const int tid  = threadIdx.x;
  const int w    = tid >> 5;        // wave id in block
  const int lane = tid & 31;
  const int hi   = lane >> 4;       // 0: lanes 0-15, 1: lanes 16-31
  const int lo   = lane & 15;
  const int b0   = (blockIdx.x * kWaves + w) * kTile;   // batch row base of tile

  // Kick off async staging of site 0 before doing anything else.
  stage_site(midT, 0, smM[0], tid);

  // ---- init edge: E[row,:] = left_tensors[q[row,0], :] ----
  for (int idx = lane; idx < kTile * kD; idx += 32) {
    const int row = idx >> 6;
    const int col = idx & (kD - 1);
    const int q0  = qn[(b0 + row) * kL];
    smE[w][idx] = leftT[q0 * kD + col];
  }

  // ---- march through the 126 middle sites, double-buffered ----
  for (int s = 0; s < kMid; ++s) {
    wait_async0();     // my chunks of site s have landed in LDS
    __syncthreads();   // everyone's chunks landed; buffer (s+1)&1 free to refill

    if (s + 1 < kMid) stage_site(midT, s + 1, smM[(s + 1) & 1], tid);

    const float* __restrict__ M = smM[s & 1];

    // q-selects for this site, issued early to hide latency under the WMMAs.
    // C/D layout: VGPR r, lane l -> row = r + 8*hi
    int qsel[8];
#pragma unroll
    for (int r = 0; r < 8; ++r)
      qsel[r] = qn[(b0 + r + 8 * hi) * kL + (s + 1)];

    v8f acc0[4], acc1[4];
    const v8f vzero = {0.f, 0.f, 0.f, 0.f, 0.f, 0.f, 0.f, 0.f};
#pragma unroll
    for (int nt = 0; nt < 4; ++nt) { acc0[nt] = vzero; acc1[nt] = vzero; }

    // K loop: 16 chunks of K=4
    for (int kk = 0; kk < kD / 4; ++kk) {
      const int kc = kk * 4 + hi * 2;   // this lane's first K column
      // A fragment (16x4 f32): lane<16 -> (M=lane, K=kc,kc+1); lane>=16 -> K+=2
      v2f a;
      a.x = smE[w][lo * kD + kc];
      a.y = smE[w][lo * kD + kc + 1];
#pragma unroll
      for (int nt = 0; nt < 4; ++nt) {
        const int nc = nt * 16 + lo;    // N column for this lane
        // B fragments (4x16 f32) for q=0 and q=1, row-major M[k][n]
        v2f bf0, bf1;
        bf0.x = M[kc * kD + nc];
        bf0.y = M[(kc + 1) * kD + nc];
        bf1.x = M[kD * kD + kc * kD + nc];
        bf1.y = M[kD * kD + (kc + 1) * kD + nc];
        // D = A*B + C   (8 args: neg_a, A, neg_b, B, c_mod, C, reuse_a, reuse_b)
        acc0[nt] = __builtin_amdgcn_wmma_f32_16x16x4_f32(
            false, a, false, bf0, (short)0, acc0[nt], false, false);
        acc1[nt] = __builtin_amdgcn_wmma_f32_16x16x4_f32(
            false, a, false, bf1, (short)0, acc1[nt], false, false);
      }
    }

    // per-row select between E*M0 and E*M1, write E back to LDS.
#pragma unroll
    for (int nt = 0; nt < 4; ++nt) {
#pragma unroll
      for (int r = 0; r < 8; ++r) {
        const int row = r + 8 * hi;
        const float v = qsel[r] ? acc1[nt][r] : acc0[nt][r];
        smE[w][row * kD + nt * 16 + lo] = v;
      }
    }
  }

  // ---- finalize: psi[row] = right[q_last] . E[row,:], out = log(complex(psi)) ----
  if (lane < kTile) {
    const int row = lane;
    const int ql  = qn[(b0 + row) * kL + (kL - 1)];
    const float* rv = rightT + ql * kD;
    float acc = 0.f;
#pragma unroll 8
    for (int c = 0; c < kD; ++c) acc += rv[c] * smE[w][row * kD + c];
    const float re = logf(fabsf(acc));
    const float im = (acc < 0.f) ? 3.14159265358979323846f : 0.f;
    out[(b0 + row) * 2 + 0] = re;
    out[(b0 + row) * 2 + 1] = im;
  }
}

extern "C" void kernel_launch(void* const* d_in, const int* in_sizes, int n_in,
                              void* d_out, int out_size, void* d_ws, size_t ws_size,
                              hipStream_t stream) {
  const int*   qn     = (const int*)d_in[0];
  const float* leftT  = (const float*)d_in[1];
  const float* rightT = (const float*)d_in[2];
  const float* midT   = (const float*)d_in[3];
  float*       out    = (float*)d_out;
  mps_logpsi_kernel<<<kBlocks, kThreads, 0, stream>>>(qn, leftT, rightT, midT, out);
}